// RBF_PINNs_87058987090244
// MI455X (gfx1250) — compile-verified
//
#include <hip/hip_runtime.h>

typedef __attribute__((ext_vector_type(2))) float v2f;
typedef __attribute__((ext_vector_type(8))) float v8f;

#define NPTS   262144
#define LOG2E  1.4426950408889634f

// ws layout (bytes):
//   0    : float sums[8] (6 used)
//   64   : v2f Atiles[4][512]  -> At_u, Ar_u, At_v, Ar_v  (16 KB)
//          per-tile flat index: tile*32 + lane
//          A 16x4 f32 layout: lane holds M=lane%16 (unit row);
//          lanes 0-15 -> {K0,K1}, lanes 16-31 -> {K2,K3}

// ---------------- kernel 0: zero sums + pack A tiles ----------------
__global__ __launch_bounds__(256) void rbf_prep(
    const float* __restrict__ mu_u, const float* __restrict__ beta_u,
    const float* __restrict__ mu_v, const float* __restrict__ beta_v,
    float* __restrict__ sums, v2f* __restrict__ At)
{
    int tid = threadIdx.x;
    if (tid < 8) sums[tid] = 0.0f;
    for (int idx = tid; idx < 2048; idx += 256) {
        int arr   = idx >> 9;         // 0:At_u 1:Ar_u 2:At_v 3:Ar_v
        int pos   = idx & 511;
        int tile  = pos >> 5;
        int lane  = pos & 31;
        int m     = lane & 15;
        int khalf = lane >> 4;
        int j     = tile * 16 + m;
        const float* mu = (arr < 2) ? mu_u   : mu_v;
        const float* be = (arr < 2) ? beta_u : beta_v;
        float m0 = mu[j], m1 = mu[256 + j];
        float mm = m0 * m0 + m1 * m1;
        float k0, k1, k2, k3;
        if ((arr & 1) == 0) {          // t-factors: t = s*r2, s = -beta*log2e
            float s = -be[j] * LOG2E;
            k0 = -2.0f * s * m0; k1 = -2.0f * s * m1; k2 = s; k3 = s * mm;
        } else {                       // r2-factors
            k0 = -2.0f * m0; k1 = -2.0f * m1; k2 = 1.0f; k3 = mm;
        }
        v2f val;
        val[0] = khalf ? k2 : k0;
        val[1] = khalf ? k3 : k1;
        At[idx] = val;
    }
}

// ---------------- helpers ----------------
__device__ __forceinline__ void ld8(const float* p, float* q) {
    float4 a = *(const float4*)p;
    float4 b = *(const float4*)(p + 4);
    q[0]=a.x; q[1]=a.y; q[2]=a.z; q[3]=a.w;
    q[4]=b.x; q[5]=b.y; q[6]=b.z; q[7]=b.w;
}

__device__ __forceinline__ void net_step(
    const v2f* __restrict__ At, const v2f* __restrict__ Ar, int lidx,
    const float* sw, const float* sc0, const float* sc1, int jb,
    v2f bvec, float& accP, float& accL)
{
    v8f zero = {};
    v2f at = At[lidx];
    v2f ar = Ar[lidx];
    // D[M=unit][N=point]: tile of t and r2 values, one point per lane
    v8f tm = __builtin_amdgcn_wmma_f32_16x16x4_f32(false, at, false, bvec,
                                                   (short)0, zero, false, false);
    v8f rm = __builtin_amdgcn_wmma_f32_16x16x4_f32(false, ar, false, bvec,
                                                   (short)0, zero, false, false);
    float wq[8], c0q[8], c1q[8];
    ld8(sw  + jb, wq);
    ld8(sc0 + jb, c0q);
    ld8(sc1 + jb, c1q);
#pragma unroll
    for (int r = 0; r < 8; ++r) {
        float phi  = __builtin_amdgcn_exp2f(tm[r]);                 // exp(-beta*r2)
        float lapw = __builtin_fmaf(rm[r], c1q[r], -c0q[r]);        // (4b^2 r2 - 4b)*W
        accP = __builtin_fmaf(phi, wq[r], accP);
        accL = __builtin_fmaf(phi, lapw, accL);
    }
}

// ---------------- kernel 1: main compute ----------------
__global__ __launch_bounds__(256) void rbf_main(
    const float* __restrict__ x,  const float* __restrict__ y,
    const float* __restrict__ u,  const float* __restrict__ v,
    const float* __restrict__ du, const float* __restrict__ dv,
    const float* __restrict__ beta_u, const float* __restrict__ W_u,
    const float* __restrict__ beta_v, const float* __restrict__ W_v,
    const float* __restrict__ dvar,  const float* __restrict__ avar,
    const float* __restrict__ bvar,  const float* __restrict__ umean,
    const float* __restrict__ vmean, const float* __restrict__ uscale,
    const float* __restrict__ vscale,
    const v2f* __restrict__ At, float* __restrict__ sums)
{
    __shared__ float sm[6 * 256];   // W_u, 4bW_u, 4b^2W_u, W_v, 4bW_v, 4b^2W_v
    int tid = threadIdx.x;
    for (int i = tid; i < 6 * 256; i += 256) {
        int which = i >> 8, j = i & 255;
        const float* be = (which < 3) ? beta_u : beta_v;
        const float* W  = (which < 3) ? W_u    : W_v;
        float b = be[j], w = W[j];
        int sel = which - ((which < 3) ? 0 : 3);
        sm[i] = (sel == 0) ? w : (sel == 1) ? 4.0f * b * w : 4.0f * b * b * w;
    }
    __syncthreads();

    const float* swu  = sm;
    const float* sc0u = sm + 256;
    const float* sc1u = sm + 512;
    const float* swv  = sm + 768;
    const float* sc0v = sm + 1024;
    const float* sc1v = sm + 1280;

    int lane  = tid & 31;
    int wave  = (blockIdx.x * blockDim.x + tid) >> 5;
    int base  = wave << 4;          // 16 points per wave
    int n     = lane & 15;
    int khalf = lane >> 4;

    float px = x[base + n];
    float py = y[base + n];

    // B 4x16 f32: lanes 0-15 hold {K0,K1} = {px,py}; lanes 16-31 {K2,K3} = {|p|^2, 1}
    v2f bvec;
    bvec[0] = khalf ? __builtin_fmaf(px, px, py * py) : px;
    bvec[1] = khalf ? 1.0f : py;

    const v2f* Atu = At;
    const v2f* Aru = At + 512;
    const v2f* Atv = At + 1024;
    const v2f* Arv = At + 1536;

    float accPu = 0.f, accLu = 0.f, accPv = 0.f, accLv = 0.f;

    for (int tile = 0; tile < 16; ++tile) {
        int lidx = tile * 32 + lane;
        int jb   = tile * 16 + khalf * 8;   // rows r -> unit j = jb + r
        net_step(Atu, Aru, lidx, swu, sc0u, sc1u, jb, bvec, accPu, accLu);
        net_step(Atv, Arv, lidx, swv, sc0v, sc1v, jb, bvec, accPv, accLv);
    }

    // merge unit-halves: lane n and lane n+16 both belong to point base+n
    float pred_u = accPu + __shfl_xor(accPu, 16, 32);
    float lap_u  = accLu + __shfl_xor(accLu, 16, 32);
    float pred_v = accPv + __shfl_xor(accPv, 16, 32);
    float lap_v  = accLv + __shfl_xor(accLv, 16, 32);

    float accs[6] = {0.f, 0.f, 0.f, 0.f, 0.f, 0.f};
    if (lane < 16) {
        float dd = dvar[0], aa = avar[0], bb = bvar[0];
        float um = umean[0], vm = vmean[0], us = uscale[0], vs = vscale[0];
        float a_us = aa / us, um_us = um / us, vm_vs = vm / vs;
        int i = base + lane;
        float uu = u[i], vv = v[i], dfu = du[i], dfv = dv[i];
        float eu = pred_u - uu; accs[0] = eu * eu;
        float ev = pred_v - vv; accs[1] = ev * ev;
        float uph = pred_u * us + um;
        float vph = pred_v * vs + vm;
        float den = 1.0f + uph * uph;
        float rpu = dd * lap_u + a_us - pred_u - um_us
                    - bb * (4.0f * (pred_u + um_us) * vph) / den;
        accs[2] = rpu * rpu;
        float rdu = lap_u - dfu; accs[3] = rdu * rdu;
        float rpv = lap_v + uph / vs - bb * uph * (pred_v + vm_vs) / den;
        accs[4] = rpv * rpv;
        float rdv = lap_v - dfv; accs[5] = rdv * rdv;
    }
#pragma unroll
    for (int k2 = 0; k2 < 6; ++k2) {
        accs[k2] += __shfl_xor(accs[k2], 1, 32);
        accs[k2] += __shfl_xor(accs[k2], 2, 32);
        accs[k2] += __shfl_xor(accs[k2], 4, 32);
        accs[k2] += __shfl_xor(accs[k2], 8, 32);
        accs[k2] += __shfl_xor(accs[k2], 16, 32);
    }
    if (lane == 0) {
#pragma unroll
        for (int k2 = 0; k2 < 6; ++k2)
            atomicAdd(&sums[k2], accs[k2]);
    }
}

// ---------------- kernel 2: finalize 7 outputs ----------------
__global__ void rbf_final(const float* __restrict__ sums, float* __restrict__ out)
{
    if (threadIdx.x == 0) {
        float inv = 1.0f / (float)NPTS;
        float lu  = sums[0] * inv;
        float lv  = sums[1] * inv;
        float pu  = sums[2] * inv;
        float duu = sums[3] * inv;
        float pv  = sums[4] * inv;
        float dvv = sums[5] * inv;
        out[0] = 0.1f * lu + 0.1f * lv + pu + duu + pv + dvv;
        out[1] = lu;  out[2] = pu;  out[3] = lv;
        out[4] = pv;  out[5] = duu; out[6] = dvv;
    }
}

extern "C" void kernel_launch(void* const* d_in, const int* in_sizes, int n_in,
                              void* d_out, int out_size, void* d_ws, size_t ws_size,
                              hipStream_t stream) {
    const float* x      = (const float*)d_in[0];
    const float* y      = (const float*)d_in[1];
    const float* u      = (const float*)d_in[2];
    const float* v      = (const float*)d_in[3];
    const float* du     = (const float*)d_in[4];
    const float* dv     = (const float*)d_in[5];
    const float* mu_u   = (const float*)d_in[6];
    const float* beta_u = (const float*)d_in[7];
    const float* W_u    = (const float*)d_in[8];
    const float* mu_v   = (const float*)d_in[9];
    const float* beta_v = (const float*)d_in[10];
    const float* W_v    = (const float*)d_in[11];
    const float* dvar   = (const float*)d_in[12];
    const float* avar   = (const float*)d_in[13];
    const float* bvar   = (const float*)d_in[14];
    const float* umean  = (const float*)d_in[15];
    const float* vmean  = (const float*)d_in[16];
    const float* uscale = (const float*)d_in[17];
    const float* vscale = (const float*)d_in[18];

    float* sums = (float*)d_ws;
    v2f*   At   = (v2f*)((char*)d_ws + 64);

    rbf_prep<<<1, 256, 0, stream>>>(mu_u, beta_u, mu_v, beta_v, sums, At);

    // 2048 blocks * 8 waves * 16 points = 262144 points
    rbf_main<<<2048, 256, 0, stream>>>(x, y, u, v, du, dv,
                                       beta_u, W_u, beta_v, W_v,
                                       dvar, avar, bvar, umean, vmean, uscale, vscale,
                                       At, sums);

    rbf_final<<<1, 32, 0, stream>>>(sums, (float*)d_out);
}